// LucyRNNCellTritonFused_89988154786160
// MI455X (gfx1250) — compile-verified
//
#include <hip/hip_runtime.h>
#include <hip/hip_bf16.h>
#include <stdint.h>

typedef __attribute__((ext_vector_type(2))) float v2f;
typedef __attribute__((ext_vector_type(8))) float v8f;
typedef __attribute__((ext_vector_type(4))) unsigned int u32x4;
typedef __attribute__((ext_vector_type(8))) int i32x8;
typedef __attribute__((ext_vector_type(4))) int i32x4;

#define USE_TDM 1           // 1: Tensor Data Mover staging; 0: async-VMEM staging

#define Bsz   32
#define Tt    2048
#define KDIN  512
#define Dd    512
#define CH    16            // channels per workgroup slice
#define TCH   16            // timesteps per chunk (WMMA M)
#define KPAD  516           // padded LDS row (516 % 64 == 4 -> conflict-free frag reads)
#define KFRAG (KDIN / 4)    // 128 k-steps of the f32 WMMA (K=4)
#define NCHUNK (Tt / TCH)

// LDS layout (floats): Z tiles first, then double-buffered A chunks.
#define ZROW   17
#define Z_OFF  0
#define ZSZ    (6 * TCH * ZROW)      // 1632 floats
#define ABUF   (TCH * KPAD)          // 8256 floats per A buffer
#define A0_OFF ZSZ
#define A1_OFF (A0_OFF + ABUF)
#define LDS_FLOATS (A1_OFF + ABUF)   // ~72.6 KB of the 320 KB WGP LDS

__device__ __forceinline__ float fast_sig(float z) {
#if __has_builtin(__builtin_amdgcn_rcpf)
    return __builtin_amdgcn_rcpf(1.0f + __expf(-z));
#else
    return 1.0f / (1.0f + __expf(-z));
#endif
}

__device__ __forceinline__ float fast_tanh(float x) {
#if __has_builtin(__builtin_amdgcn_tanhf)
    return __builtin_amdgcn_tanhf(x);   // native gfx1250 V_TANH_F32
#else
    return tanhf(x);
#endif
}

#if USE_TDM
// One TDM descriptor moves a [16 x 512] f32 tile into LDS.
// pad_enable inserts 2 dwords after every 256 dwords streamed -> each 512-dword
// row occupies 516 dwords in LDS, with cols >= 256 shifted by +2.
__device__ __forceinline__ void tdm_load_chunk(uint32_t lds_byte, const float* gptr)
{
    const uint64_t ga = (uint64_t)(uintptr_t)gptr;
    u32x4 g0;
    g0[0] = 1u;                                   // count=1 (valid descriptor)
    g0[1] = lds_byte;                             // lds_addr (bytes)
    g0[2] = (uint32_t)ga;                         // global_addr[31:0]
    g0[3] = (uint32_t)(ga >> 32) | 0x80000000u;   // global_addr[56:32] | type=2<<30

    i32x8 g1;
    g1[0] = (2 << 16)          // data_size = 4 bytes
          | (1 << 20)          // pad_enable
          | (7 << 22)          // pad_interval = 256 dwords
          | (1 << 25);         // pad_amount   = 2 dwords
    g1[1] = (int)(512u << 16);         // tensor_dim0[15:0]=512 (bits 63:48)
    g1[2] = 0;                         // tensor_dim0 hi / tensor_dim1 lo
    g1[3] = (int)(1u | (512u << 16));  // tensor_dim1[31:16]=1 (65536 rows), tile_dim0=512
    g1[4] = 16;                        // tile_dim1=16 rows, tile_dim2=0 (2D)
    g1[5] = 512;                       // tensor_dim0_stride lo32 = 512 elements
    g1[6] = 0;                         // dim0_stride hi / dim1_stride lo (unused, 2D)
    g1[7] = 0;

    const i32x4 gz  = {0, 0, 0, 0};               // groups 2/3: unused dims
    const i32x8 gz8 = {0, 0, 0, 0, 0, 0, 0, 0};   // 6-arg form: extra group, zero
    __builtin_amdgcn_tensor_load_to_lds(g0, g1, gz, gz, gz8, 0);
}
#else
// Fallback: ASYNCcnt-tracked per-wave DMA of one x chunk into LDS.
__device__ __forceinline__ void stage_chunk_async(const float* __restrict__ x,
                                                  const float* lds_base,
                                                  int abuf_off, int b, int t0, int tid)
{
    for (int idx = tid; idx < TCH * (KDIN / 4); idx += 192) {
        const int m  = idx >> 7;
        const int c4 = idx & 127;
        const float* gp = &x[((size_t)b * Tt + (t0 + m)) * KDIN + c4 * 4];
        const uint32_t laddr =
            (uint32_t)(uintptr_t)(const void*)&lds_base[abuf_off + m * KPAD + c4 * 4];
        asm volatile("global_load_async_to_lds_b128 %0, %1, off"
                     :: "v"(laddr), "v"(gp) : "memory");
    }
}
__device__ __forceinline__ void wait_async_all() {
    asm volatile("s_wait_asynccnt 0" ::: "memory");
}
#endif

__launch_bounds__(192, 1)
__global__ void lucy_fused_kernel(const float* __restrict__ x,     // [B,T,KDIN]
                                  const float* __restrict__ h0,    // [B,D]
                                  const float* __restrict__ s0,    // [B,D]
                                  const float* __restrict__ W,     // [6D,KDIN]
                                  const float* __restrict__ bias,  // [6D]
                                  float* __restrict__ out,         // [B,T,D]
                                  float* __restrict__ sT)          // [B,D]
{
    extern __shared__ float lds[];
    const int tid  = threadIdx.x;
    const int gate = tid >> 5;      // wave id == gate id (0..5)
    const int lane = tid & 31;
    const int b    = blockIdx.y;
    const int d0   = blockIdx.x * CH;

    // Fragment addressing per ISA 32-bit 16x4 layout:
    //   lane l holds op[row = l%16, K = (l/16)*2 + v], v in {0,1}
    const int mrow  = lane & 15;
    const int khalf = (lane >> 4) * 2;

    // ---- This wave's entire W gate-tile in 256 persistent VGPRs ----
    v2f wfrag[KFRAG];
    {
        const float* wrow = W + (size_t)(gate * Dd + d0 + mrow) * KDIN + khalf;
        #pragma unroll
        for (int kk = 0; kk < KFRAG; ++kk)
            wfrag[kk] = *(const v2f*)(wrow + kk * 4);
    }
    const float biasv = bias[gate * Dd + d0 + mrow];

    float h = 0.f, s = 0.f;
    if (tid < CH) {
        h = h0[b * Dd + d0 + tid];
        s = s0[b * Dd + d0 + tid];
    }

    float* Zg = &lds[Z_OFF + gate * TCH * ZROW];

    const uint32_t a0_byte = (uint32_t)(uintptr_t)(const void*)&lds[A0_OFF];
    const uint32_t a1_byte = (uint32_t)(uintptr_t)(const void*)&lds[A1_OFF];
    const float* xb = x + (size_t)b * Tt * KDIN;

    // ---- Prologue: stage chunk 0 ----
#if USE_TDM
    if (gate == 0) {
        tdm_load_chunk(a0_byte, xb);
        __builtin_amdgcn_s_wait_tensorcnt((short)0);
    }
#else
    stage_chunk_async(x, lds, A0_OFF, b, 0, tid);
    wait_async_all();
#endif
    __syncthreads();

    for (int tc = 0; tc < NCHUNK; ++tc) {
        const int t0      = tc * TCH;
        const int cur_off = (tc & 1) ? A1_OFF : A0_OFF;

        // Kick off DMA for the next chunk; overlaps with the WMMA loop below.
        if (tc + 1 < NCHUNK) {
#if USE_TDM
            if (gate == 0)
                tdm_load_chunk((tc & 1) ? a0_byte : a1_byte,
                               xb + (size_t)(t0 + TCH) * KDIN);
#else
            stage_chunk_async(x, lds, (tc & 1) ? A0_OFF : A1_OFF, b, t0 + TCH, tid);
#endif
        }

        // ---- Z_gate[16t x 16ch] = A(16x512) * Wgate^T, B operands from VGPRs ----
        // TDM padding lands mid-row at col 256 -> +2 dword shift for kk >= 64.
        v8f acc = {};
        const float* Ar = &lds[cur_off + mrow * KPAD + khalf];
        #pragma unroll
        for (int kk = 0; kk < KFRAG / 2; ++kk) {
            const v2f av = *(const v2f*)(Ar + kk * 4);
            acc = __builtin_amdgcn_wmma_f32_16x16x4_f32(
                false, av, false, wfrag[kk], (short)0, acc, false, false);
        }
        #pragma unroll
        for (int kk = KFRAG / 2; kk < KFRAG; ++kk) {
#if USE_TDM
            const v2f av = *(const v2f*)(Ar + kk * 4 + 2);
#else
            const v2f av = *(const v2f*)(Ar + kk * 4);
#endif
            acc = __builtin_amdgcn_wmma_f32_16x16x4_f32(
                false, av, false, wfrag[kk], (short)0, acc, false, false);
        }
        // C/D layout: lane l, vgpr r -> (M = (l/16)*8 + r, N = l%16)
        #pragma unroll
        for (int r = 0; r < 8; ++r) {
            const int m = (lane >> 4) * 8 + r;
            Zg[m * ZROW + mrow] = acc[r] + biasv;
        }
        __syncthreads();   // Z tiles visible to scan wave

        // ---- Sequential 16-step recurrence (wave 0, 16 lanes) ----
        if (tid < CH) {
            const float* Z = &lds[Z_OFF];
            #pragma unroll
            for (int m = 0; m < TCH; ++m) {
                const int zi = m * ZROW + tid;
                const float z0 = Z[0 * TCH * ZROW + zi];
                const float z1 = Z[1 * TCH * ZROW + zi];
                const float z2 = Z[2 * TCH * ZROW + zi];
                const float z3 = Z[3 * TCH * ZROW + zi];
                const float z4 = Z[4 * TCH * ZROW + zi];
                const float z5 = Z[5 * TCH * ZROW + zi];
                const float sn = fast_sig(z0) * s + fast_sig(z1) * fast_tanh(z2);
                const float hn = fast_sig(z3) * h + fast_sig(z4) * fast_tanh(z5 + sn);
                s = sn; h = hn;
                out[((size_t)b * Tt + (t0 + m)) * Dd + d0 + tid] = hn;
            }
        }

        // Next chunk's DMA must have landed before any wave reads the flipped buffer.
#if USE_TDM
        if (gate == 0)
            __builtin_amdgcn_s_wait_tensorcnt((short)0);
#else
        wait_async_all();
#endif
        __syncthreads();
    }

    if (tid < CH) sT[b * Dd + d0 + tid] = s;
}

extern "C" void kernel_launch(void* const* d_in, const int* in_sizes, int n_in,
                              void* d_out, int out_size, void* d_ws, size_t ws_size,
                              hipStream_t stream) {
    const float* x    = (const float*)d_in[0];
    const float* h0   = (const float*)d_in[1];
    const float* s0   = (const float*)d_in[2];
    const float* W    = (const float*)d_in[3];
    const float* bias = (const float*)d_in[4];

    float* out = (float*)d_out;                       // [B,T,D]
    float* sT  = out + (size_t)Bsz * Tt * Dd;         // then [B,D]

    const size_t ldsBytes = (size_t)LDS_FLOATS * sizeof(float);
    (void)hipFuncSetAttribute((const void*)lucy_fused_kernel,
                              hipFuncAttributeMaxDynamicSharedMemorySize,
                              (int)ldsBytes);

    dim3 grid(Dd / CH, Bsz);   // 32 channel-slices x 32 batches = 1024 WGs
    lucy_fused_kernel<<<grid, 192, ldsBytes, stream>>>(x, h0, s0, W, bias, out, sT);
}